// ConvNdFunc_77318001262715
// MI455X (gfx1250) — compile-verified
//
#include <hip/hip_runtime.h>

// 3x3 same-padding conv, 4096x4096 fp32, via V_WMMA_F32_16X16X4_F32.
// Memory-bound (AI ~2.25 FLOP/B, ~5.8us HBM floor at 23.3 TB/s):
//  - stage 18x68 halo slabs in LDS once (1.16x read amplification),
//  - run the 9-tap conv as A(16x18 input) x B(18x16 Toeplitz-of-weights)
//    on the matrix pipe; each wave produces a 16x64 strip (4 accumulators,
//    60 WMMAs).
//  - NEW: the 15 banded-Toeplitz B chunks are built once per block in LDS
//    and loaded into registers with paired ds loads, removing the ~180
//    v_cndmask/v_cmp ops per wave that round 1 spent rebuilding B inline.

typedef float v2f __attribute__((ext_vector_type(2)));
typedef float v8f __attribute__((ext_vector_type(8)));

#define IMG_N   4096
#define TILE_M  16      // output rows per wave
#define TILE_N  64      // output cols per wave (4 WMMA n-tiles)
#define SLAB_R  18      // TILE_M + 2 halo rows
#define SLAB_C  68      // 66 valid cols (TILE_N + 2 halo) + 2 zero pad (K=16..19 chunk)
#define WPB     8       // waves per block
#define TOEP_K  20      // K extent of Toeplitz table (18 valid + 2 zero)

__global__ __launch_bounds__(256) void conv3x3_wmma_kernel(
    const float* __restrict__ x,
    const float* __restrict__ wgt,
    const float* __restrict__ bias,
    float* __restrict__ out)
{
    __shared__ float slab[WPB][SLAB_R][SLAB_C];           // 39168 B
    __shared__ float toep[3][TOEP_K][16];                 //  3840 B

    const int tid  = threadIdx.x;
    const int lane = tid & 31;
    const int wid  = tid >> 5;

    // 256 tile-rows x 64 tile-cols of 16x64 strips; consecutive waves take
    // horizontally adjacent strips for L2 locality.
    const int tile = blockIdx.x * WPB + wid;              // 0..16383
    const int trow = tile >> 6;
    const int tcol = tile & 63;
    const int row0 = trow * TILE_M;
    const int col0 = tcol * TILE_N;

    // ---- build banded Toeplitz weight table, cooperatively (once/block) ----
    // toep[dy][k][n] = w[3*dy + (k-n)] when 0 <= k-n <= 2, else 0
    for (int i = tid; i < 3 * TOEP_K * 16; i += 256) {
        const int dy  = i / (TOEP_K * 16);
        const int rem = i - dy * (TOEP_K * 16);
        const int k   = rem >> 4;
        const int n   = rem & 15;
        const int d   = k - n;
        float v = 0.0f;
        if (d >= 0 && d < 3) v = wgt[3 * dy + d];
        (&toep[0][0][0])[i] = v;
    }

    // ---- stage input slab (zero halo at borders, zero pad cols 66..67) ----
    float* s = &slab[wid][0][0];
    #pragma unroll
    for (int r = 0; r < SLAB_R; ++r) {
        const int gr = row0 - 1 + r;
        const bool rok = (gr >= 0) && (gr < IMG_N);
        const float* xrow = x + (long)gr * IMG_N;
        for (int c = lane; c < SLAB_C; c += 32) {
            const int gc = col0 - 1 + c;
            float v = 0.0f;
            if (rok && (c < 66) && (gc >= 0) && (gc < IMG_N)) v = xrow[gc];
            s[r * SLAB_C + c] = v;
        }
    }
    __syncthreads();   // slab + toep visible; EXEC all-1s afterwards (WMMA req)

    const float b0 = bias[0];

    // fp32 WMMA operand layouts (ISA 7.12.2):
    //  A 16x4 : lanes 0-15 hold K=0,1 ; lanes 16-31 hold K=2,3  (m = lane&15)
    //  B 4x16 : lanes 0-15 hold K=0,1 ; lanes 16-31 hold K=2,3  (n = lane&15)
    //  C 16x16: vgpr r = row r (lanes 0-15) / row r+8 (lanes 16-31)
    const int m   = lane & 15;
    const int n   = lane & 15;
    const int kb2 = (lane >> 4) << 1;          // 0 or 2

    // ---- load all 15 loop-invariant B chunks into registers ----
    v2f btab[15];
    #pragma unroll
    for (int dy = 0; dy < 3; ++dy) {
        #pragma unroll
        for (int kk = 0; kk < 5; ++kk) {
            const float* tp = &toep[dy][kk * 4 + kb2][n];
            v2f bm;
            bm.x = tp[0];      // row kb
            bm.y = tp[16];     // row kb+1 (64B apart -> ds_load_2addr_b32)
            btab[dy * 5 + kk] = bm;
        }
    }

    v8f acc0 = {0.f,0.f,0.f,0.f,0.f,0.f,0.f,0.f};
    v8f acc1 = acc0, acc2 = acc0, acc3 = acc0;

    #pragma unroll
    for (int dy = 0; dy < 3; ++dy) {
        const float* srow = &slab[wid][m + dy][0];
        #pragma unroll
        for (int kk = 0; kk < 5; ++kk) {
            const int kb = kk * 4 + kb2;
            const v2f bmat = btab[dy * 5 + kk];
            #pragma unroll
            for (int j = 0; j < 4; ++j) {
                v2f amat;                       // ds_load_2addr_b64 pairs
                amat.x = srow[16 * j + kb];
                amat.y = srow[16 * j + kb + 1];
                v8f* accp = (j == 0) ? &acc0 : (j == 1) ? &acc1
                          : (j == 2) ? &acc2 : &acc3;
                *accp = __builtin_amdgcn_wmma_f32_16x16x4_f32(
                    false, amat, false, bmat, (short)0, *accp, false, false);
            }
        }
    }

    // ---- store: vgpr r -> rows (mb+r), 16 consecutive cols per half-wave ----
    const int mb = (lane >> 4) << 3;            // 0 or 8
    float* orow = out + (long)(row0 + mb) * IMG_N + col0 + n;
    #pragma unroll
    for (int r = 0; r < 8; ++r) {
        orow[(long)r * IMG_N +  0] = acc0[r] + b0;
        orow[(long)r * IMG_N + 16] = acc1[r] + b0;
        orow[(long)r * IMG_N + 32] = acc2[r] + b0;
        orow[(long)r * IMG_N + 48] = acc3[r] + b0;
    }
}

extern "C" void kernel_launch(void* const* d_in, const int* in_sizes, int n_in,
                              void* d_out, int out_size, void* d_ws, size_t ws_size,
                              hipStream_t stream) {
    const float* x  = (const float*)d_in[0];
    const float* w  = (const float*)d_in[1];
    const float* b  = (const float*)d_in[2];
    float* out      = (float*)d_out;

    const int tiles  = (IMG_N / TILE_M) * (IMG_N / TILE_N);  // 16384
    const int blocks = tiles / WPB;                           // 2048
    conv3x3_wmma_kernel<<<dim3(blocks), dim3(256), 0, stream>>>(x, w, b, out);
}